// ConvDCT_15126874816879
// MI455X (gfx1250) — compile-verified
//
#include <hip/hip_runtime.h>

// ConvDCT on MI455X (gfx1250):
//   K  = DCT2(pad3x3(w))           -> 9-term direct sum, store-bound (256MB)
//   X  = DCT2(x)                   -> two separable 32-pt passes (FMA)
//   R  = batched GEMM over c       -> V_WMMA_F32_16X16X4_F32 (hot spot, 8.6 GFLOP)
//   out= crop(IDCT2(R))            -> two separable passes (FMA)
// Layouts chosen so the WMMA kernel's A/B loads are per-lane b64 with zero
// intra-wave redundancy: X=[hw][n][c], K=[hw][f][c], R=[hw][n][f].
// k_mix uses a 16(n)x64(f) tile per wave: 1 A load amortized over 4 WMMAs.

typedef __attribute__((ext_vector_type(2))) float v2f;
typedef __attribute__((ext_vector_type(8))) float v8f;

#define NN 64
#define CH 256
#define FO 256
#define HW 1024   // 32*32 spectral positions

// ---------------------------------------------------------------- DCT mats
__global__ __launch_bounds__(256) void k_build_mats(float* __restrict__ D,
                                                    float* __restrict__ Mi) {
  int t = blockIdx.x * 256 + threadIdx.x;        // 1024 entries, 32x32 each
  int r = t >> 5, c = t & 31;
  const float PI = 3.14159265358979323846f;
  // forward: D[k][i] = 2*cos(pi*k*(2i+1)/64)
  D[t] = 2.0f * __cosf(PI * (float)r * (2.0f * (float)c + 1.0f) / 64.0f);
  // inverse: Mi[h][k] = cos(pi*k*(2h+1)/64) * ((k==0?0.5:1)/32)
  float wgt = (c == 0 ? 0.5f : 1.0f) / 32.0f;
  Mi[t] = __cosf(PI * (float)c * (2.0f * (float)r + 1.0f) / 64.0f) * wgt;
}

// ------------------------------------------------- spectral kernel (3x3 -> 32x32)
// Ks[(k*32+l)*F*C + f*C + c] = sum_{a<3,b<3} D[k][a]*W[f][c][a][b]*D[l][b]
__global__ __launch_bounds__(256) void k_kspec(const float* __restrict__ Wt,
                                               const float* __restrict__ D,
                                               float* __restrict__ Ks) {
  size_t gid = (size_t)blockIdx.x * 256 + threadIdx.x;
  int c  = (int)(gid & 255);
  int f  = (int)((gid >> 8) & 255);
  int hw = (int)(gid >> 16);
  int k = hw >> 5, l = hw & 31;
  float dk0 = D[k*32+0], dk1 = D[k*32+1], dk2 = D[k*32+2];
  float dl0 = D[l*32+0], dl1 = D[l*32+1], dl2 = D[l*32+2];
  const float* wp = Wt + (size_t)(f * 256 + c) * 9;
  float r0 = wp[0]*dl0 + wp[1]*dl1 + wp[2]*dl2;
  float r1 = wp[3]*dl0 + wp[4]*dl1 + wp[5]*dl2;
  float r2 = wp[6]*dl0 + wp[7]*dl1 + wp[8]*dl2;
  Ks[gid] = dk0*r0 + dk1*r1 + dk2*r2;
}

// ------------------------------------------------- forward DCT pass 1 (over h)
// T[n][c][k][w] = sum_h D[k][h] * x[n][c][h][w]
__global__ __launch_bounds__(256) void k_dct_h(const float* __restrict__ x,
                                               const float* __restrict__ D,
                                               float* __restrict__ T) {
  size_t gid = (size_t)blockIdx.x * 256 + threadIdx.x;
  int w = (int)(gid & 31);
  int k = (int)((gid >> 5) & 31);
  size_t nc = gid >> 10;
  const float* xp = x + nc * 1024 + (size_t)w;
  const float* dp = D + k * 32;
  float acc = 0.0f;
#pragma unroll
  for (int h = 0; h < 32; ++h) acc += dp[h] * xp[(size_t)h * 32];
  T[gid] = acc;
}

// ------------------------------------------------- forward DCT pass 2 (over w)
// X[(k*32+l)][n][c] = sum_w T[n][c][k][w] * D[l][w]
__global__ __launch_bounds__(256) void k_dct_w(const float* __restrict__ T,
                                               const float* __restrict__ D,
                                               float* __restrict__ X) {
  size_t gid = (size_t)blockIdx.x * 256 + threadIdx.x;
  int c  = (int)(gid & 255);
  int n  = (int)((gid >> 8) & 63);
  int hw = (int)(gid >> 14);
  int k = hw >> 5, l = hw & 31;
  const float* tp = T + (((size_t)(n * 256 + c) * 32) + (size_t)k) * 32;
  const float* dp = D + l * 32;
  float acc = 0.0f;
#pragma unroll
  for (int w = 0; w < 32; ++w) acc += tp[w] * dp[w];
  X[gid] = acc;
}

// ------------------------------------------------- spectral mixing (WMMA f32)
// R[hw][n][f] = sum_c X[hw][n][c] * Ks[hw][f][c]
// Batched GEMM: 1024 batches of M=64(n), N=256(f), K=256(c).
// One wave32 computes a 16x64 (n,f) macro-tile: 4 accumulators, the A
// fragment is loaded once per K-step and fed to 4 WMMAs (4 wmma : 5 loads).
__global__ __launch_bounds__(256) void k_mix(const float* __restrict__ X,
                                             const float* __restrict__ Ks,
                                             float* __restrict__ R) {
  const int hw   = blockIdx.x;             // 0..1023
  const int wave = threadIdx.x >> 5;       // 0..7
  const int lane = threadIdx.x & 31;
  const int gw   = blockIdx.y * 8 + wave;  // 0..15
  const int tn = gw & 3;                   // n-tile: rows tn*16 .. tn*16+15
  const int fg = gw >> 2;                  // f-group: cols fg*64 .. fg*64+63
  const int m    = lane & 15;              // M (A) / N (B) index within tile
  const int koff = (lane >> 4) << 1;       // lanes 16..31 hold K=2,3 halves

  const float* Arow = X  + ((size_t)hw * NN + (size_t)(tn * 16 + m)) * CH + koff;
  const float* Brow = Ks + ((size_t)hw * FO + (size_t)(fg * 64 + m)) * CH + koff;

  // Pre-stage operand rows into cache (global_prefetch_b8).
#pragma unroll
  for (int p = 0; p < 256; p += 64) {
    __builtin_prefetch(Arow + p, 0, 1);
    __builtin_prefetch(Brow + p, 0, 1);
    __builtin_prefetch(Brow + 16 * CH + p, 0, 1);
    __builtin_prefetch(Brow + 32 * CH + p, 0, 1);
    __builtin_prefetch(Brow + 48 * CH + p, 0, 1);
  }

  v8f acc0 = {}, acc1 = {}, acc2 = {}, acc3 = {};
#pragma unroll 4
  for (int k0 = 0; k0 < 256; k0 += 4) {
    v2f a  = *(const v2f*)(Arow + k0);             // A: 16x4 f32, lane=M, pair=K
    v2f b0 = *(const v2f*)(Brow + k0);             // B: 4x16 f32, f-tile 0
    v2f b1 = *(const v2f*)(Brow + k0 + 16 * CH);   // f-tile 1 (+16KB imm offset)
    v2f b2 = *(const v2f*)(Brow + k0 + 32 * CH);   // f-tile 2
    v2f b3 = *(const v2f*)(Brow + k0 + 48 * CH);   // f-tile 3
    acc0 = __builtin_amdgcn_wmma_f32_16x16x4_f32(false, a, false, b0, (short)0, acc0, false, false);
    acc1 = __builtin_amdgcn_wmma_f32_16x16x4_f32(false, a, false, b1, (short)0, acc1, false, false);
    acc2 = __builtin_amdgcn_wmma_f32_16x16x4_f32(false, a, false, b2, (short)0, acc2, false, false);
    acc3 = __builtin_amdgcn_wmma_f32_16x16x4_f32(false, a, false, b3, (short)0, acc3, false, false);
  }

  // C/D layout: VGPR r -> M = r + 8*(lane>=16), N = lane&15
  float* out = R + ((size_t)hw * NN + (size_t)(tn * 16 + 8 * (lane >> 4))) * FO
                 + (size_t)(fg * 64 + (lane & 15));
#pragma unroll
  for (int r = 0; r < 8; ++r) {
    out[(size_t)r * FO +  0] = acc0[r];
    out[(size_t)r * FO + 16] = acc1[r];
    out[(size_t)r * FO + 32] = acc2[r];
    out[(size_t)r * FO + 48] = acc3[r];
  }
}

// ------------------------------------------------- inverse DCT pass 1 (over k)
// S[h][l][n][f] = sum_k Mi[h][k] * R[(k*32+l)][n][f]   (h < 30)
__global__ __launch_bounds__(256) void k_idct_h(const float* __restrict__ R,
                                                const float* __restrict__ Mi,
                                                float* __restrict__ S) {
  size_t gid = (size_t)blockIdx.x * 256 + threadIdx.x;
  int f = (int)(gid & 255);
  int n = (int)((gid >> 8) & 63);
  int l = (int)((gid >> 14) & 31);
  int h = (int)(gid >> 19);
  const float* rp = R + (size_t)l * 16384 + (size_t)n * 256 + (size_t)f;
  const float* mp = Mi + h * 32;
  float acc = 0.0f;
#pragma unroll
  for (int k = 0; k < 32; ++k) acc += mp[k] * rp[(size_t)k * 524288];
  S[gid] = acc;
}

// ------------------------------------------------- inverse DCT pass 2 + crop
// out[n][f][h][w] = sum_l S[h][l][n][f] * Mi[w][l]   (h,w < 30)
__global__ __launch_bounds__(256) void k_idct_w(const float* __restrict__ S,
                                                const float* __restrict__ Mi,
                                                float* __restrict__ out) {
  size_t gid = (size_t)blockIdx.x * 256 + threadIdx.x;
  int w = (int)(gid % 30);
  size_t r = gid / 30;
  int h = (int)(r % 30);
  size_t nf = r / 30;                    // n*256+f
  const float* sp = S + (size_t)h * 524288 + nf;
  const float* mp = Mi + w * 32;
  float acc = 0.0f;
#pragma unroll
  for (int l = 0; l < 32; ++l) acc += mp[l] * sp[(size_t)l * 16384];
  out[gid] = acc;
}

extern "C" void kernel_launch(void* const* d_in, const int* in_sizes, int n_in,
                              void* d_out, int out_size, void* d_ws, size_t ws_size,
                              hipStream_t stream) {
  const float* x  = (const float*)d_in[0];  // [64,256,32,32]
  const float* wt = (const float*)d_in[1];  // [256,256,3,3]
  float* out = (float*)d_out;               // [64,256,30,30]
  float* ws  = (float*)d_ws;

  // Workspace layout (floats):
  float* D  = ws;                                   //   1024  (4 KB)
  float* Mi = ws + 1024;                            //   1024  (4 KB)
  float* Ks = ws + 2048;                            // 1024*256*256  (256 MB)
  float* X  = Ks + (size_t)HW * FO * CH;            // 1024*64*256   (64 MB), reused as S
  float* T  = X  + (size_t)HW * NN * CH;            // 64*256*32*32  (64 MB), reused as R

  k_build_mats<<<4,      256, 0, stream>>>(D, Mi);
  k_kspec     <<<262144, 256, 0, stream>>>(wt, D, Ks);     // 1024*256*256 / 256
  k_dct_h     <<<65536,  256, 0, stream>>>(x, D, T);       // 16.7M / 256
  k_dct_w     <<<65536,  256, 0, stream>>>(T, D, X);
  k_mix       <<<dim3(1024, 2), 256, 0, stream>>>(X, Ks, T /* = R */);
  k_idct_h    <<<61440,  256, 0, stream>>>(T /*R*/, Mi, X /* = S */); // 30*32*64*256/256
  k_idct_w    <<<57600,  256, 0, stream>>>(X /*S*/, Mi, out);         // 64*256*30*30/256
}